// TemporalConsistencyLoss_4320737099938
// MI455X (gfx1250) — compile-verified
//
#include <hip/hip_runtime.h>
#include <hip/hip_bf16.h>

typedef __attribute__((ext_vector_type(2))) float v2f;
typedef __attribute__((ext_vector_type(8))) float v8f;

#define EPS_CS   1e-8f
#define N_TRAJ   128
#define T_STEP   64
#define D_DIM    128
#define LDS_STR  132   // padded row stride in floats; 132*4 = 528 B (16B aligned)

// Grid: N_TRAJ*2 blocks; each block handles a 32x64 sub-block of one trajectory's
// 64x64 cosine-similarity block. Block = 256 threads = 8 wave32s, one 16x16 WMMA
// tile per wave (2 tile-rows x 4 tile-cols).
__global__ __launch_bounds__(256)
void tcl_block_kernel(const float* __restrict__ p,
                      const float* __restrict__ z,
                      float* __restrict__ partial) {
    __shared__ float lds_p[32 * LDS_STR];
    __shared__ float lds_z[64 * LDS_STR];
    __shared__ float pnorm[32];
    __shared__ float znorm[64];
    __shared__ float wavesum[8];

    const int t     = threadIdx.x;
    const int blk   = blockIdx.x;
    const int traj  = blk >> 1;
    const int half  = blk & 1;
    const int prow0 = traj * T_STEP + half * 32;   // 32 P rows for this block
    const int zrow0 = traj * T_STEP;               // all 64 Z rows of the trajectory

    // Hint the front of both tiles into cache (global_prefetch_b8 on gfx1250).
    __builtin_prefetch(p + (size_t)prow0 * D_DIM, 0, 3);
    __builtin_prefetch(z + (size_t)zrow0 * D_DIM, 0, 3);

    // ---- Stage P (32x128 f32) into LDS via float4 (1024 f4 -> 4 per thread) ----
#pragma unroll
    for (int i = 0; i < 4; ++i) {
        const int g   = t + 256 * i;   // float4 index
        const int row = g >> 5;        // 32 float4 per row
        const int c4  = g & 31;
        const float4 v = *reinterpret_cast<const float4*>(
            p + (size_t)(prow0 + row) * D_DIM + c4 * 4);
        *reinterpret_cast<float4*>(&lds_p[row * LDS_STR + c4 * 4]) = v;
    }
    // ---- Stage Z (64x128 f32) into LDS (2048 f4 -> 8 per thread) ----
#pragma unroll
    for (int i = 0; i < 8; ++i) {
        const int g   = t + 256 * i;
        const int row = g >> 5;
        const int c4  = g & 31;
        const float4 v = *reinterpret_cast<const float4*>(
            z + (size_t)(zrow0 + row) * D_DIM + c4 * 4);
        *reinterpret_cast<float4*>(&lds_z[row * LDS_STR + c4 * 4]) = v;
    }
    __syncthreads();

    // ---- Row L2 norms: threads 0..31 -> P rows, 32..95 -> Z rows ----
    if (t < 96) {
        const bool isz = (t >= 32);
        const int  row = isz ? (t - 32) : t;
        const float* src = isz ? &lds_z[row * LDS_STR] : &lds_p[row * LDS_STR];
        float s = 0.0f;
#pragma unroll 8
        for (int k = 0; k < D_DIM; ++k) { const float x = src[k]; s += x * x; }
        const float n = sqrtf(s);
        if (isz) znorm[row] = n; else pnorm[row] = n;
    }
    __syncthreads();

    // ---- fp32 WMMA: one 16x16 tile per wave, K = 128 in steps of 4 ----
    const int wave   = t >> 5;
    const int lane   = t & 31;
    const int half16 = lane >> 4;   // lane group selects K pair {0,1} vs {2,3}
    const int lm     = lane & 15;   // matrix row (A) / col (B) within the tile
    const int m0     = (wave >> 2) * 16;   // 0 or 16   (P tile row)
    const int n0     = (wave & 3)  * 16;   // 0..48     (Z tile col)

    const float* arow = &lds_p[(m0 + lm) * LDS_STR + 2 * half16];
    const float* brow = &lds_z[(n0 + lm) * LDS_STR + 2 * half16];

    v8f acc = {};
#pragma unroll
    for (int k = 0; k < D_DIM; k += 4) {
        const v2f a = *reinterpret_cast<const v2f*>(arow + k);  // ds_load_b64
        const v2f b = *reinterpret_cast<const v2f*>(brow + k);  // ds_load_b64
        // D = A(16x4 f32) x B(4x16 f32) + C  -> v_wmma_f32_16x16x4_f32
        acc = __builtin_amdgcn_wmma_f32_16x16x4_f32(
            /*neg_a=*/false, a, /*neg_b=*/false, b,
            /*c_mod=*/(short)0, acc, /*reuse_a=*/false, /*reuse_b=*/false);
    }

    // ---- Epilogue: divide by max(|p_i||z_j|, eps) and reduce ----
    float lsum = 0.0f;
#pragma unroll
    for (int r = 0; r < 8; ++r) {
        const int i = m0 + half16 * 8 + r;   // local P row (0..31)
        const int j = n0 + lm;               // local Z row (0..63)
        lsum += acc[r] / fmaxf(pnorm[i] * znorm[j], EPS_CS);
    }
    // wave32 tree reduction
#pragma unroll
    for (int off = 16; off > 0; off >>= 1)
        lsum += __shfl_down(lsum, off, 32);
    if (lane == 0) wavesum[wave] = lsum;
    __syncthreads();

    if (t == 0) {
        float s = 0.0f;
#pragma unroll
        for (int w = 0; w < 8; ++w) s += wavesum[w];
        partial[blk] = s;
    }
}

// Deterministic fixed-order reduction of the 256 per-block partials.
__global__ __launch_bounds__(256)
void tcl_finalize_kernel(const float* __restrict__ partial,
                         float* __restrict__ out) {
    __shared__ float red[256];
    const int t = threadIdx.x;
    red[t] = partial[t];
    __syncthreads();
#pragma unroll
    for (int s = 128; s > 0; s >>= 1) {
        if (t < s) red[t] += red[t + s];
        __syncthreads();
    }
    if (t == 0) {
        const float scale = 1.0f / ((float)(N_TRAJ * T_STEP) * (float)T_STEP);
        out[0] = -red[0] * scale;   // loss = -mean_i( mean_j in-block logits )
    }
}

extern "C" void kernel_launch(void* const* d_in, const int* in_sizes, int n_in,
                              void* d_out, int out_size, void* d_ws, size_t ws_size,
                              hipStream_t stream) {
    const float* p = (const float*)d_in[0];
    const float* z = (const float*)d_in[1];
    // d_in[2] (done) is provably dead in the reference: w == temporal_mask.
    float* partial = (float*)d_ws;       // 256 floats of scratch
    float* out     = (float*)d_out;      // single f32 loss

    tcl_block_kernel<<<N_TRAJ * 2, 256, 0, stream>>>(p, z, partial);
    tcl_finalize_kernel<<<1, 256, 0, stream>>>(partial, out);
}